// MLP_learner_53541062312462
// MI455X (gfx1250) — compile-verified
//
#include <hip/hip_runtime.h>
#include <hip/hip_bf16.h>

typedef __attribute__((ext_vector_type(2))) float v2f;
typedef __attribute__((ext_vector_type(8))) float v8f;

#define NN 12288
#define DD 128
#define KK 31

// ---------------------------------------------------------------------------
// K0: h = relu(X@W1^T+b1); h = h@W2^T+b2; hn = h / max(||h||,1e-12)
// One 128-thread workgroup per row. W1/W2 (64KB each) stay hot in L2.
// ---------------------------------------------------------------------------
__global__ void k_mlp_norm(const float* __restrict__ X,
                           const float* __restrict__ W1,
                           const float* __restrict__ b1,
                           const float* __restrict__ W2,
                           const float* __restrict__ b2,
                           float* __restrict__ hn) {
    __shared__ float xs[DD];
    __shared__ float h1[DD];
    __shared__ float red[DD];
    const int row = blockIdx.x;
    const int j   = threadIdx.x;

    xs[j] = X[(size_t)row * DD + j];
    __syncthreads();

    float acc = b1[j];
    #pragma unroll 8
    for (int k = 0; k < DD; ++k) acc += xs[k] * W1[j * DD + k];
    h1[j] = fmaxf(acc, 0.0f);
    __syncthreads();

    float acc2 = b2[j];
    #pragma unroll 8
    for (int k = 0; k < DD; ++k) acc2 += h1[k] * W2[j * DD + k];

    red[j] = acc2 * acc2;
    __syncthreads();
    for (int s = 64; s > 0; s >>= 1) {
        if (j < s) red[j] += red[j + s];
        __syncthreads();
    }
    const float inv = 1.0f / fmaxf(sqrtf(red[0]), 1e-12f);
    hn[(size_t)row * DD + j] = acc2 * inv;
}

// ---------------------------------------------------------------------------
// K1: sim = hn @ hn^T via V_WMMA_F32_16X16X4_F32 with fused streaming top-31.
// 4 waves per WG; wave w owns rows [blk*64 + w*16, +16). Columns streamed in
// 16-wide tiles; K=128 -> 32 WMMA per tile, B loads batched 8-at-a-time.
//
// Register budget (wave32): A frags 64 + C 8 + B batch 16 + temps ~ 120 VGPRs.
// Top-31 (val,idx) lists live in per-lane private LDS slabs (stride 31 dwords
// lane-to-lane: odd stride -> conflict-free across 64 banks); hot path only
// compares against a register-cached running min, insertion is a rare
// unrolled static-offset LDS RMW pass.
// ---------------------------------------------------------------------------
__global__ void __launch_bounds__(128, 1)
k_sim_topk(const float* __restrict__ hn,
           float* __restrict__ topkV,
           int*   __restrict__ topkI) {
    __shared__ float simLds[4][16 * 16];
    __shared__ float listV[4][32][KK];
    __shared__ int   listI[4][32][KK];

    const int lane = threadIdx.x & 31;
    const int wave = threadIdx.x >> 5;
    const int lp   = lane & 15;
    const int half = lane >> 4;
    const int row0 = blockIdx.x * 64 + wave * 16;

    // Preload all A fragments for this wave's 16 rows (16x128 -> 32 frags).
    v2f a[32];
    const float* arow = hn + (size_t)(row0 + lp) * DD + 2 * half;
    #pragma unroll
    for (int kk = 0; kk < 32; ++kk)
        a[kk] = *(const v2f*)(arow + 4 * kk);

    // Init this lane's top-31 list in LDS; cache the running min in a VGPR.
    float* Lv = &listV[wave][lane][0];
    int*   Li = &listI[wave][lane][0];
    #pragma unroll
    for (int s = 0; s < KK; ++s) { Lv[s] = -__builtin_inff(); Li[s] = 0; }
    float vmin = -__builtin_inff();

    const int tr = lane & 15;   // row this lane tracks for top-k
    const int ch = lane >> 4;   // which 8-column half of each tile

    for (int col0 = 0; col0 < NN; col0 += 16) {
        v8f c = {0.f, 0.f, 0.f, 0.f, 0.f, 0.f, 0.f, 0.f};
        const float* brow = hn + (size_t)(col0 + lp) * DD + 2 * half;
        #pragma unroll
        for (int kb = 0; kb < 4; ++kb) {
            v2f b[8];
            #pragma unroll
            for (int i = 0; i < 8; ++i)
                b[i] = *(const v2f*)(brow + 4 * (kb * 8 + i));
            #pragma unroll
            for (int i = 0; i < 8; ++i)
                c = __builtin_amdgcn_wmma_f32_16x16x4_f32(
                        false, a[kb * 8 + i], false, b[i], (short)0, c,
                        false, false);
        }

        // Stage 16x16 tile to this wave's LDS slab (row-major).
        #pragma unroll
        for (int r = 0; r < 8; ++r)
            simLds[wave][(r + 8 * half) * 16 + lp] = c[r];
        __syncthreads();

        // Streaming top-31 update: lane (tr,ch) scans cols ch*8..ch*8+7.
        #pragma unroll
        for (int q = 0; q < 8; ++q) {
            float v  = simLds[wave][tr * 16 + ch * 8 + q];
            int   id = col0 + ch * 8 + q;
            if (v > vmin) {
                float cv = v; int ci = id;
                #pragma unroll
                for (int s = 0; s < KK; ++s) {
                    float ov = Lv[s]; int oi = Li[s];
                    bool  gt = cv > ov;
                    float nv = gt ? cv : ov;
                    int   ni = gt ? ci : oi;
                    Lv[s] = nv; Li[s] = ni;
                    cv = gt ? ov : cv; ci = gt ? oi : ci;
                    if (s == KK - 1) vmin = nv;
                }
            }
        }
        __syncthreads();
    }

    // Merge each row's two half-lists -> global top-31 for the row.
    // (Lists already in LDS: lane tr holds half 0, lane 16+tr holds half 1.)
    if (ch == 0) {
        int pa = 0, pb = 0;
        const int grow = row0 + tr;
        for (int k = 0; k < KK; ++k) {
            float va = (pa < KK) ? listV[wave][tr][pa]      : -__builtin_inff();
            float vb = (pb < KK) ? listV[wave][16 + tr][pb] : -__builtin_inff();
            bool  ta = va >= vb;
            float v  = ta ? va : vb;
            int   id = ta ? listI[wave][tr][pa] : listI[wave][16 + tr][pb];
            if (ta) ++pa; else ++pb;
            topkV[(size_t)grow * KK + k] = v;
            topkI[(size_t)grow * KK + k] = id;
        }
    }
}

// ---------------------------------------------------------------------------
// K2: out = zeros; out[row, idx_k] = relu(val_k). One WG per row; float4
// zero-fill (coalesced), then in-block scatter after syncthreads.
// ---------------------------------------------------------------------------
__global__ void k_emit(const float* __restrict__ topkV,
                       const int*   __restrict__ topkI,
                       float* __restrict__ out) {
    const int row = blockIdx.x;
    const int t   = threadIdx.x;
    float4* orow = (float4*)(out + (size_t)row * NN);
    const float4 z = {0.f, 0.f, 0.f, 0.f};
    #pragma unroll
    for (int i = 0; i < 12; ++i)          // 12288/4 = 3072 = 256 * 12
        orow[t + 256 * i] = z;
    __syncthreads();
    if (t < KK) {
        float v  = topkV[(size_t)row * KK + t];
        int   id = topkI[(size_t)row * KK + t];
        out[(size_t)row * NN + id] = fmaxf(v, 0.0f);
    }
}

// ---------------------------------------------------------------------------
extern "C" void kernel_launch(void* const* d_in, const int* in_sizes, int n_in,
                              void* d_out, int out_size, void* d_ws, size_t ws_size,
                              hipStream_t stream) {
    (void)in_sizes; (void)n_in; (void)out_size; (void)ws_size;
    const float* X  = (const float*)d_in[0];
    const float* W1 = (const float*)d_in[1];
    const float* b1 = (const float*)d_in[2];
    const float* W2 = (const float*)d_in[3];
    const float* b2 = (const float*)d_in[4];
    float* out = (float*)d_out;

    // ws layout: topkV | topkI | hn   (all offsets 8-byte aligned)
    char* ws = (char*)d_ws;
    float* topkV = (float*)ws;                               // N*31 floats
    int*   topkI = (int*)(ws + (size_t)NN * KK * 4);         // N*31 ints
    float* hn    = (float*)(ws + (size_t)NN * KK * 8);       // N*128 floats

    k_mlp_norm<<<NN, DD, 0, stream>>>(X, W1, b1, W2, b2, hn);
    k_sim_topk<<<NN / 64, 128, 0, stream>>>(hn, topkV, topkI);
    k_emit<<<NN, 256, 0, stream>>>(topkV, topkI, out);
}